// HaHCost_43353399886066
// MI455X (gfx1250) — compile-verified
//
#include <hip/hip_runtime.h>

typedef __attribute__((ext_vector_type(2))) float v2f;
typedef __attribute__((ext_vector_type(8))) float v8f;

#define NCH   512
#define KTOP  52
#define NBOT  460          /* 512 - 52 */
#define HW    1024         /* 32*32  */
#define TILE  32
#define NWG   2048         /* 64 batches * (1024/32) tiles */
#define NPOS  65536.0f

// XOR swizzle: conflict-free for row-major writes (lane==col) and for
// column reads (lane==row low bits), since (l&1)<<5 | ((c^l)&31) is a
// bijection over 32 lanes into 64 banks.
__device__ __forceinline__ int lds_idx(int row, int col) {
  return (row << 5) | (col ^ (row & 31));
}

__global__ __launch_bounds__(256) void hah_topk_kernel(const float* __restrict__ x,
                                                       float* __restrict__ partial) {
  __shared__ float lds[NCH * TILE];   // 64 KB, xor-swizzled ReLU'd tile
  const int tid = threadIdx.x;
  const int wg  = blockIdx.x;
  const int b   = wg >> 5;            // batch
  const int hw0 = (wg & 31) << 5;     // first of 32 spatial positions
  const size_t base = (size_t)b * NCH * HW + (size_t)hw0;

  // Stage 512x32 tile: fully coalesced 128B row segments per wave.
  for (int i = tid; i < NCH * TILE; i += 256) {
    const int row = i >> 5;
    const int col = i & 31;
    const float v = x[base + (size_t)row * HW + col];
    lds[lds_idx(row, col)] = fmaxf(v, 0.0f);
  }
  __syncthreads();

  const int wave = tid >> 5;
  const int lane = tid & 31;
  float acc = 0.0f;

  // Each wave32 owns 4 columns (positions); lane j holds channels lane, lane+32, ...
  for (int cc = 0; cc < 4; ++cc) {
    const int col = (wave << 2) + cc;
    float    v[16];
    unsigned u[16];
#pragma unroll
    for (int j = 0; j < 16; ++j) {
      v[j] = lds[lds_idx(lane + (j << 5), col)];
      u[j] = __float_as_uint(v[j]);          // nonneg floats: u32 order == float order
    }

    // Total ReLU sum S over 512 channels (butterfly -> uniform in wave).
    float s = 0.0f;
#pragma unroll
    for (int j = 0; j < 16; ++j) s += v[j];
#pragma unroll
    for (int off = 16; off >= 1; off >>= 1) s += __shfl_xor(s, off, 32);

    // Exact radix select: largest t with count(u >= t) >= K  ==  K-th largest value.
    // Ballot counts keep cnt/t wave-uniform (SALU), co-issuing with the v_cmps.
    unsigned t = 0u;
    for (int bit = 30; bit >= 0; --bit) {
      const unsigned cand = t | (1u << bit);
      int cnt = 0;
#pragma unroll
      for (int j = 0; j < 16; ++j)
        cnt += (int)__popcll(__ballot(u[j] >= cand));
      if (cnt >= KTOP) t = cand;
    }

    // Strictly-greater sum + count; pad remaining slots with tie value t.
    float sg = 0.0f;
    int   cg = 0;
#pragma unroll
    for (int j = 0; j < 16; ++j) {
      const bool g = (u[j] > t);
      sg += g ? v[j] : 0.0f;
      cg += (int)__popcll(__ballot(g));      // wave-uniform running count
    }
#pragma unroll
    for (int off = 16; off >= 1; off >>= 1) sg += __shfl_xor(sg, off, 32);

    const float topSum = sg + (float)(KTOP - cg) * __uint_as_float(t);
    acc += topSum / (float)KTOP - (s - topSum) / (float)NBOT;
  }

  // Per-workgroup partial (reuse LDS after all reads complete).
  __syncthreads();
  if (lane == 0) lds[wave] = acc;
  __syncthreads();
  if (tid == 0) {
    float p = 0.0f;
    for (int w = 0; w < 8; ++w) p += lds[w];
    partial[wg] = p;
  }
}

// Deterministic final reduction of 2048 partials via fp32 WMMA:
// D = A(16x4 data) * B(4x16 ones) + C accumulates row sums; with B all-ones,
// any bijective packing of the 64-value chunk into A is summed exactly.
__global__ __launch_bounds__(32) void hah_reduce_kernel(const float* __restrict__ partial,
                                                        float* __restrict__ out) {
  const int lane = threadIdx.x;   // single wave32, EXEC all ones for WMMA
  v8f c = {};
  v2f ones;
  ones.x = 1.0f;
  ones.y = 1.0f;

  for (int chunk = 0; chunk < NWG / 64; ++chunk) {
    const float* p = partial + chunk * 64;
    v2f a;
    a.x = p[lane * 2 + 0];
    a.y = p[lane * 2 + 1];
    // 8 args: (neg_a, A, neg_b, B, c_mod, C, reuse_a, reuse_b)
    c = __builtin_amdgcn_wmma_f32_16x16x4_f32(false, a, false, ones,
                                              (short)0, c, false, false);
  }

  // lanes 0-15 hold D rows 0..7 (cols = lane), lanes 16-31 rows 8..15.
  float s8 = c[0] + c[1] + c[2] + c[3] + c[4] + c[5] + c[6] + c[7];
  s8 += __shfl_xor(s8, 16, 32);   // rows 0..7 + rows 8..15 at same column
  if (lane == 0) out[0] = s8 * (1.0f / NPOS);
}

extern "C" void kernel_launch(void* const* d_in, const int* in_sizes, int n_in,
                              void* d_out, int out_size, void* d_ws, size_t ws_size,
                              hipStream_t stream) {
  const float* x  = (const float*)d_in[0];
  float*       ws = (float*)d_ws;          // 2048 fp32 partials
  float*       o  = (float*)d_out;

  hah_topk_kernel<<<dim3(NWG), dim3(256), 0, stream>>>(x, ws);
  hah_reduce_kernel<<<dim3(1), dim3(32), 0, stream>>>(ws, o);
}